// Transformer_7464653161080
// MI455X (gfx1250) — compile-verified
//
#include <hip/hip_runtime.h>

typedef float v2f __attribute__((ext_vector_type(2)));
typedef float v4f __attribute__((ext_vector_type(4)));
typedef float v8f __attribute__((ext_vector_type(8)));

#define HEADS 8
#define DKDIM 32
#define FEAT  (HEADS * DKDIM)   // 256 floats per node

__global__ void zero_f32_kernel(float* __restrict__ p, int n) {
    int i = blockIdx.x * blockDim.x + threadIdx.x;
    if (i < n) p[i] = 0.0f;
}

// One block = 16 edges; wave w handles head w for those 16 edges.
// A = k[src[e0..e15]][h][:]  (16x32, f32, row-striped per ISA 7.12.2)
// B = q[dst[e0..e15]][h][:]^T (32x16)
// C = A*B via 8x V_WMMA_F32_16X16X4_F32; scores are diag(C).
__global__ __launch_bounds__(256) void edge_attn_wmma_kernel(
    const float* __restrict__ q, const float* __restrict__ k,
    const float* __restrict__ v, const int* __restrict__ src,
    const int* __restrict__ dst, float* __restrict__ wv,
    float* __restrict__ z, int E) {

    const int lane = threadIdx.x & 31;
    const int h    = threadIdx.x >> 5;     // wave id == head (H==8)
    const int em   = lane & 15;            // edge-in-tile this lane serves
    const int half = lane >> 4;            // lane group 0: dims {4c,4c+1}, 1: {4c+2,4c+3}

    long e = (long)blockIdx.x * 16 + em;
    const bool valid = (e < (long)E);
    if (e >= (long)E) e = (long)E - 1;     // branchless clamp: WMMA needs EXEC all-1s

    const int s = src[e];
    const int d = dst[e];

    const float* __restrict__ krow = k + (size_t)s * FEAT + h * DKDIM;
    const float* __restrict__ qrow = q + (size_t)d * FEAT + h * DKDIM;

    // --- 8 chained f32 WMMAs over K = 32 (4 per instruction) ---
    v8f acc = {};
#pragma unroll
    for (int c = 0; c < 8; ++c) {
        // A 16x4 chunk: lane em holds dims 4c+2*half, 4c+2*half+1 of edge em's k-row
        v2f a = *(const v2f*)(krow + 4 * c + 2 * half);
        // B 4x16 chunk: same per-lane dim pair of edge em's q-row (column N=em)
        v2f b = *(const v2f*)(qrow + 4 * c + 2 * half);
        acc = __builtin_amdgcn_wmma_f32_16x16x4_f32(
            /*neg_a=*/false, a, /*neg_b=*/false, b,
            /*c_mod=*/(short)0, acc, /*reuse_a=*/false, /*reuse_b=*/false);
    }

    // --- extract diag(C): (i,i) lives in lane i (i<8, vgpr i) or lane 16+i (i>=8, vgpr i-8)
    const int idx = lane & 7;  // meaningful for lanes 0-7 and 24-31
    float dv = acc[0];
    dv = (idx == 1) ? acc[1] : dv;
    dv = (idx == 2) ? acc[2] : dv;
    dv = (idx == 3) ? acc[3] : dv;
    dv = (idx == 4) ? acc[4] : dv;
    dv = (idx == 5) ? acc[5] : dv;
    dv = (idx == 6) ? acc[6] : dv;
    dv = (idx == 7) ? acc[7] : dv;
    const int src_lane = em + ((em >= 8) ? 16 : 0);
    float sc = __shfl(dv, src_lane, 32);   // ds_bpermute_b32, wave32

    // --- exp(clip(score / sqrt(32), -5, 5)) ---
    const float inv_scale = 0.1767766952966369f;  // 1/sqrt(32)
    sc = sc * inv_scale;
    sc = fminf(5.0f, fmaxf(-5.0f, sc));
    const float score = __expf(sc);

    // --- scatter: lanes em and em+16 split the 32-dim wv row (16 dims each) ---
    if (valid) {
        const float* __restrict__ vrow = v  + (size_t)s * FEAT + h * DKDIM + half * 16;
        float* __restrict__       wrow = wv + (size_t)d * FEAT + h * DKDIM + half * 16;
#pragma unroll
        for (int j = 0; j < 16; j += 4) {
            v4f vv = *(const v4f*)(vrow + j);       // global_load_b128
            atomicAdd(wrow + j + 0, vv.x * score);  // global_atomic_add_f32 (no rtn)
            atomicAdd(wrow + j + 1, vv.y * score);
            atomicAdd(wrow + j + 2, vv.z * score);
            atomicAdd(wrow + j + 3, vv.w * score);
        }
        if (half == 0)
            atomicAdd(z + (size_t)d * HEADS + h, score);
    }
}

extern "C" void kernel_launch(void* const* d_in, const int* in_sizes, int n_in,
                              void* d_out, int out_size, void* d_ws, size_t ws_size,
                              hipStream_t stream) {
    const float* q  = (const float*)d_in[0];
    const float* k  = (const float*)d_in[1];
    const float* v  = (const float*)d_in[2];
    const int* srci = (const int*)d_in[3];
    const int* dsti = (const int*)d_in[4];

    const int E       = in_sizes[3];
    const int n_nodes = in_sizes[0] / FEAT;

    float* wv = (float*)d_out;                       // [N, H, DK]
    float* z  = wv + (size_t)n_nodes * FEAT;         // [N, H]

    // Outputs are poisoned 0xAA: zero wv and z before accumulating.
    zero_f32_kernel<<<(out_size + 255) / 256, 256, 0, stream>>>((float*)d_out, out_size);

    const int tiles = (E + 15) / 16;                 // 16 edges per block
    edge_attn_wmma_kernel<<<tiles, 256, 0, stream>>>(q, k, v, srci, dsti, wv, z, E);
}